// DoubleIntCell_30425548325160
// MI455X (gfx1250) — compile-verified
//
#include <hip/hip_runtime.h>
#include <math.h>
#include <stdint.h>

// Problem constants from the reference: T=64, B=2048, U=1024.
#define TT 64
#define BB 2048
#define UU 1024
#define RSTRIDE (2 * UU)  // recurrent_kernel row stride (U x 2U, row-major)
#define KC 32             // K rows staged in LDS per pipeline stage
#define LDSPITCH 72       // 64 data dwords + 8 pad dwords -> lanes 0-15 / 16-31 bank-disjoint

typedef float v2f __attribute__((ext_vector_type(2)));
typedef float v8f __attribute__((ext_vector_type(8)));

#if defined(__AMDGCN__) && __has_builtin(__builtin_amdgcn_tensor_load_to_lds)
#define USE_TDM 1
typedef unsigned int tdm_v4u __attribute__((ext_vector_type(4)));
typedef int tdm_v8i __attribute__((ext_vector_type(8)));
typedef int tdm_v4i __attribute__((ext_vector_type(4)));

// Issue a TDM load of a 64-col x KC-row f32 tile of R (row stride 2048 elems)
// into LDS at lds_byte_addr, padded 8 dwords per 64 (-> pitch LDSPITCH dwords).
__device__ __forceinline__ void tdm_load_tile(unsigned lds_byte_addr,
                                              const float* gtile) {
    unsigned long long ga = (unsigned long long)(uintptr_t)gtile;
    tdm_v4u g0;
    g0[0] = 1u;                                   // count=1 (valid user D#)
    g0[1] = lds_byte_addr;                        // LDS dest (bytes)
    g0[2] = (unsigned)ga;                         // global_addr[31:0]
    g0[3] = (unsigned)((ga >> 32) & 0x01FFFFFFu)  // global_addr[56:32]
            | (2u << 30);                         // type = 2 ("image")
    tdm_v8i g1;
    g1[0] = (int)((2u << 16)      // data_size = 4 bytes
                  | (1u << 20)    // pad_enable
                  | (5u << 22)    // pad_interval: every 64 dwords
                  | (7u << 25));  // pad_amount: 8 dwords
    g1[1] = (int)(((unsigned)RSTRIDE & 0xFFFFu) << 16);          // tensor_dim0 lo16
    g1[2] = (int)((((unsigned)RSTRIDE >> 16) & 0xFFFFu)          // tensor_dim0 hi16
                  | (((unsigned)UU & 0xFFFFu) << 16));           // tensor_dim1 lo16
    g1[3] = (int)((((unsigned)UU >> 16) & 0xFFFFu) | (64u << 16)); // dim1 hi | tile_dim0=64
    g1[4] = (int)(unsigned)KC;   // tile_dim1 = KC, tile_dim2 = 0 (2D)
    g1[5] = (int)RSTRIDE;        // tensor_dim0_stride[31:0]
    g1[6] = 0;                   // stride hi / dim1_stride lo
    g1[7] = 0;
    tdm_v4i z4 = {0, 0, 0, 0};                // groups 2/3 unused for 2D tile
    tdm_v8i z8 = {0, 0, 0, 0, 0, 0, 0, 0};    // extra group in 6-arg form
    __builtin_amdgcn_tensor_load_to_lds(g0, g1, z4, z4, z8, 0);
}
#endif

// ---------------------------------------------------------------------------
// Per-timestep fused kernel (exact-fp32 WMMA path):
//   X = h_old @ [Rf | Rc]  ;  f = sigmoid(X_f + x1*Kf + bf)
//   c = f*h_old + (1-f)*tanh(X_c + x1*Kc + bc) -> h_new ; y[b] += c.w
// Block = 8 waves sharing one 64-column N-strip; each wave owns 16 rows.
// R tiles are staged into LDS by the Tensor Data Mover (double buffered).
// ---------------------------------------------------------------------------
__global__ __launch_bounds__(256) void step_gemm_cell(
    const float* __restrict__ hold,  // B x U
    float*       __restrict__ hnew,  // B x U
    const float* __restrict__ R,     // U x 2U
    const float* __restrict__ Kb,    // 2U
    const float* __restrict__ bias,  // 2U
    const float* __restrict__ w,     // U
    const float* __restrict__ x1s,   // B
    float*       __restrict__ y)     // B (pre-zeroed accumulator)
{
    __shared__ float lds_f[2][KC * LDSPITCH];
    __shared__ float lds_c[2][KC * LDSPITCH];

    const int lane = threadIdx.x & 31;
    const int wid  = threadIdx.x >> 5;
    const int bm   = blockIdx.x >> 4;        // 16 row-blocks of 128
    const int bn   = blockIdx.x & 15;        // 16 col-strips of 64
    const int m0   = bm * 128 + wid * 16;
    const int u0   = bn * 64;
    const int half = lane >> 4;
    const int l16  = lane & 15;

    v8f accf[4] = {};
    v8f accc[4] = {};

    // A operand: lane holds h[m0+l16][k + 2*half .. +1]
    const float* aptr = hold + (size_t)(m0 + l16) * UU + 2 * half;

    // One K-chunk of WMMAs fed from the LDS stage.
    auto compute_chunk = [&](const float* Lf, const float* Lc) {
#pragma unroll 4
        for (int kk = 0; kk < KC; kk += 4) {
            v2f a = *(const v2f*)aptr;
            aptr += 4;
            const int kb = (kk + 2 * half) * LDSPITCH;
#pragma unroll
            for (int nt = 0; nt < 4; ++nt) {
                const int col = nt * 16 + l16;
                v2f bf_, bc_;
                bf_[0] = Lf[kb + col];
                bf_[1] = Lf[kb + LDSPITCH + col];
                bc_[0] = Lc[kb + col];
                bc_[1] = Lc[kb + LDSPITCH + col];
                accf[nt] = __builtin_amdgcn_wmma_f32_16x16x4_f32(
                    false, a, false, bf_, (short)0, accf[nt], false, false);
                accc[nt] = __builtin_amdgcn_wmma_f32_16x16x4_f32(
                    false, a, false, bc_, (short)0, accc[nt], false, false);
            }
        }
    };

#ifdef USE_TDM
    const float* rf0 = R + u0;        // forget-gate columns
    const float* rc0 = R + UU + u0;   // candidate columns
    if (wid == 0) {                   // prologue: stage chunk 0 into buffer 0
        tdm_load_tile((unsigned)(uintptr_t)&lds_f[0][0], rf0);
        tdm_load_tile((unsigned)(uintptr_t)&lds_c[0][0], rc0);
    }
    for (int ic = 0; ic < UU / KC; ++ic) {
        if (wid == 0) __builtin_amdgcn_s_wait_tensorcnt(0);  // chunk ic resident
        __syncthreads();
        if (wid == 0 && ic + 1 < UU / KC) {  // stage chunk ic+1 into other buffer
            const int nb = (ic + 1) & 1;
            const float* gf = rf0 + (size_t)(ic + 1) * KC * RSTRIDE;
            tdm_load_tile((unsigned)(uintptr_t)&lds_f[nb][0], gf);
            tdm_load_tile((unsigned)(uintptr_t)&lds_c[nb][0], gf + UU);
        }
        compute_chunk(lds_f[ic & 1], lds_c[ic & 1]);
    }
#else
    for (int ic = 0; ic < UU / KC; ++ic) {
        const int k0 = ic * KC;
        __syncthreads();
        for (int i = threadIdx.x; i < KC * 64; i += 256) {
            const int r = i >> 6, c = i & 63;
            lds_f[0][r * LDSPITCH + c] = R[(size_t)(k0 + r) * RSTRIDE + u0 + c];
            lds_c[0][r * LDSPITCH + c] = R[(size_t)(k0 + r) * RSTRIDE + UU + u0 + c];
        }
        __syncthreads();
        compute_chunk(lds_f[0], lds_c[0]);
    }
#endif

    // Per-column constants for this wave's 64 columns.
    float kf[4], kc[4], bfv[4], bcv[4], wv[4];
#pragma unroll
    for (int nt = 0; nt < 4; ++nt) {
        const int u = u0 + nt * 16 + l16;
        kf[nt]  = Kb[u];
        kc[nt]  = Kb[UU + u];
        bfv[nt] = bias[u];
        bcv[nt] = bias[UU + u];
        wv[nt]  = w[u];
    }

    // C/D layout: VGPR r = row (m0+r) for lanes 0-15, (m0+r+8) for lanes 16-31.
#pragma unroll
    for (int r = 0; r < 8; ++r) {
        const int row  = m0 + r + 8 * half;
        const float x1r = x1s[row];
        float p = 0.0f;
#pragma unroll
        for (int nt = 0; nt < 4; ++nt) {
            const int u = u0 + nt * 16 + l16;
            float xf = accf[nt][r] + x1r * kf[nt] + bfv[nt];
            float xc = accc[nt][r] + x1r * kc[nt] + bcv[nt];
            float fg = 1.0f / (1.0f + __expf(-xf));
            float hv = hold[(size_t)row * UU + u];
            float cv = fg * hv + (1.0f - fg) * tanhf(xc);
            hnew[(size_t)row * UU + u] = cv;
            p += cv * wv[nt];
        }
#pragma unroll
        for (int off = 8; off >= 1; off >>= 1)  // reduce within each 16-lane half
            p += __shfl_xor(p, off, 32);
        if (l16 == 0) atomicAdd(&y[row], p);
    }
}

// Per-step state advance: x1' = x1 + x2 ; x2' = x2 + inp*(c@w) ; out = x1'.
__global__ __launch_bounds__(256) void step_update(
    const float* __restrict__ inp_t,
    float* __restrict__ x1s, float* __restrict__ x2s,
    float* __restrict__ y,   float* __restrict__ out_t)
{
    int b = blockIdx.x * blockDim.x + threadIdx.x;
    if (b < BB) {
        float yv  = y[b];
        float x1  = x1s[b];
        float x2  = x2s[b];
        float x1n = x1 + x2;
        x1s[b] = x1n;
        x2s[b] = x2 + inp_t[b] * yv;
        out_t[b] = x1n;
        y[b] = 0.0f;  // re-arm accumulator for next step
    }
}

__global__ __launch_bounds__(256) void init_state(
    const float* __restrict__ x1_0, const float* __restrict__ x2_0,
    float* __restrict__ x1s, float* __restrict__ x2s, float* __restrict__ y)
{
    int b = blockIdx.x * blockDim.x + threadIdx.x;
    if (b < BB) {
        x1s[b] = x1_0[b];
        x2s[b] = x2_0[b];
        y[b]   = 0.0f;
    }
}

extern "C" void kernel_launch(void* const* d_in, const int* in_sizes, int n_in,
                              void* d_out, int out_size, void* d_ws, size_t ws_size,
                              hipStream_t stream) {
    const float* inputs = (const float*)d_in[0]; // T x B
    const float* x1_0   = (const float*)d_in[1]; // B
    const float* x2_0   = (const float*)d_in[2]; // B
    const float* c0     = (const float*)d_in[3]; // B x U
    const float* Kb     = (const float*)d_in[4]; // 2U
    const float* R      = (const float*)d_in[5]; // U x 2U
    const float* bias   = (const float*)d_in[6]; // 2U
    const float* w      = (const float*)d_in[7]; // U
    float* out = (float*)d_out;

    float* ws  = (float*)d_ws;
    float* h0  = ws;                       // B*U
    float* h1  = ws + (size_t)BB * UU;     // B*U
    float* x1s = ws + 2 * (size_t)BB * UU; // B
    float* x2s = x1s + BB;                 // B
    float* y   = x2s + BB;                 // B

    // h(0) = c0 (state rebuilt on every call; graph replays the whole sequence).
    (void)hipMemcpyAsync(h0, c0, sizeof(float) * (size_t)BB * UU,
                         hipMemcpyDeviceToDevice, stream);
    init_state<<<(BB + 255) / 256, 256, 0, stream>>>(x1_0, x2_0, x1s, x2s, y);

    for (int t = 0; t < TT; ++t) {
        const float* hold = (t & 1) ? h1 : h0;
        float*       hnew = (t & 1) ? h0 : h1;
        step_gemm_cell<<<256, 256, 0, stream>>>(hold, hnew, R, Kb, bias, w, x1s, y);
        step_update<<<(BB + 255) / 256, 256, 0, stream>>>(
            inputs + (size_t)t * BB, x1s, x2s, y, out + (size_t)t * BB);
    }
}